// Net_62792421867575
// MI455X (gfx1250) — compile-verified
//
#include <hip/hip_runtime.h>
#include <hip/hip_bf16.h>
#include <math.h>

typedef float v2f __attribute__((ext_vector_type(2)));
typedef float v8f __attribute__((ext_vector_type(8)));

#define N_NODES 100000
#define F_IN    512
#define HID     16
#define NCLS    10

// ---------------------------------------------------------------- utilities
__global__ void zero_f32(float* __restrict__ p, int n) {
    int i = blockIdx.x * blockDim.x + threadIdx.x;
    if (i < n) p[i] = 0.0f;
}

__global__ void deg_count(const int* __restrict__ dst, float* __restrict__ deg, int e) {
    int i = blockIdx.x * blockDim.x + threadIdx.x;
    if (i < e) atomicAdd(&deg[dst[i]], 1.0f);
}

__global__ void finalize_dinv(float* __restrict__ deg, int n) {
    int i = blockIdx.x * blockDim.x + threadIdx.x;
    if (i < n) deg[i] = rsqrtf(deg[i] + 1.0f);   // +1 self-loop; deg>0 always
}

// ------------------------------------------- layer-1 projection: X @ W1 (WMMA)
// One wave32 per 16-row tile. A: 16x4 f32 fragment = v2f/lane
// (lanes 0-15: K=0,1 ; lanes 16-31: K=2,3). B from LDS. C/D: v8f.
__global__ __launch_bounds__(256) void gemm_x_w1(const float* __restrict__ X,
                                                 const float* __restrict__ W,
                                                 float* __restrict__ H,
                                                 int nTiles) {
    __shared__ float LW[F_IN * HID];                 // 32 KB
    for (int i = threadIdx.x; i < F_IN * HID; i += 256) LW[i] = W[i];
    __syncthreads();

    const int wave = threadIdx.x >> 5;
    const int lane = threadIdx.x & 31;
    const int tile = blockIdx.x * 8 + wave;
    if (tile >= nTiles) return;                      // wave-uniform: EXEC stays all-1s

    const int mn = lane & 15;                        // row of A frag == col of B frag
    const int kb = (lane >> 4) << 1;                 // 0 or 2
    const float* __restrict__ xrow = X + (size_t)(tile * 16 + mn) * F_IN + kb;

    v8f c = {0.f, 0.f, 0.f, 0.f, 0.f, 0.f, 0.f, 0.f};
    for (int k0 = 0; k0 < F_IN; k0 += 4) {
        __builtin_prefetch(xrow + k0 + 32, 0, 1);    // global_prefetch_b8, stream-ahead
        v2f a = *(const v2f*)(xrow + k0);
        v2f b;
        b.x = LW[(k0 + kb)     * HID + mn];
        b.y = LW[(k0 + kb + 1) * HID + mn];
        c = __builtin_amdgcn_wmma_f32_16x16x4_f32(false, a, false, b,
                                                  (short)0, c, false, false);
    }

    const int mbase = (lane >> 4) * 8;               // VGPR v -> row v (+8 for hi lanes)
    float* __restrict__ hout = H + (size_t)tile * 16 * HID;
#pragma unroll
    for (int v = 0; v < 8; ++v)
        hout[(size_t)(v + mbase) * HID + mn] = c[v];
}

// ------------------------------------------- layer-2 projection: H @ W2 (WMMA)
// K = 16 (4 WMMA k-steps); W2 zero-padded 16x16 in LDS so cols >= 10 are zero.
__global__ __launch_bounds__(256) void gemm_h_w2(const float* __restrict__ Hb,
                                                 const float* __restrict__ W2,
                                                 float* __restrict__ P,
                                                 int nTiles) {
    __shared__ float LW[HID * 16];                   // 1 KB, padded
    {
        int k = threadIdx.x >> 4, n = threadIdx.x & 15;   // 256 threads = 16x16
        LW[threadIdx.x] = (n < NCLS) ? W2[k * NCLS + n] : 0.0f;
    }
    __syncthreads();

    const int wave = threadIdx.x >> 5;
    const int lane = threadIdx.x & 31;
    const int tile = blockIdx.x * 8 + wave;
    if (tile >= nTiles) return;

    const int mn = lane & 15;
    const int kb = (lane >> 4) << 1;
    const float* __restrict__ hrow = Hb + (size_t)(tile * 16 + mn) * HID + kb;

    v8f c = {0.f, 0.f, 0.f, 0.f, 0.f, 0.f, 0.f, 0.f};
#pragma unroll
    for (int k0 = 0; k0 < HID; k0 += 4) {
        v2f a = *(const v2f*)(hrow + k0);
        v2f b;
        b.x = LW[(k0 + kb)     * 16 + mn];
        b.y = LW[(k0 + kb + 1) * 16 + mn];
        c = __builtin_amdgcn_wmma_f32_16x16x4_f32(false, a, false, b,
                                                  (short)0, c, false, false);
    }

    const int mbase = (lane >> 4) * 8;
    float* __restrict__ pout = P + (size_t)tile * 16 * 16;
#pragma unroll
    for (int v = 0; v < 8; ++v)
        pout[(size_t)(v + mbase) * 16 + mn] = c[v];
}

// ------------------------------------------------ edge scatter (gather+atomic)
template <int F, int LDA, int LDB>
__global__ __launch_bounds__(256) void scatter_edges(const int* __restrict__ src,
                                                     const int* __restrict__ dst,
                                                     const float* __restrict__ dinv,
                                                     const float* __restrict__ A,
                                                     float* __restrict__ B, int e) {
    int t = blockIdx.x * blockDim.x + threadIdx.x;
    if (t >= e) return;
    const int s = src[t];
    const int d = dst[t];
    const float w = dinv[s] * dinv[d];
    const float* __restrict__ as = A + (size_t)s * LDA;
    float* __restrict__ bd = B + (size_t)d * LDB;
#pragma unroll
    for (int f = 0; f < F; ++f) atomicAdd(&bd[f], as[f] * w);
}

// ----------------------------------------- layer-1 epilogue: self-loop+bias+ReLU
__global__ __launch_bounds__(256) void relu_bias_self(const float* __restrict__ A,
                                                      float* __restrict__ B,
                                                      const float* __restrict__ dinv,
                                                      const float* __restrict__ b1,
                                                      int n) {
    int i = blockIdx.x * blockDim.x + threadIdx.x;
    if (i >= n) return;
    const float d2 = dinv[i] * dinv[i];
#pragma unroll
    for (int f = 0; f < HID; ++f) {
        float v = B[(size_t)i * HID + f] + A[(size_t)i * HID + f] * d2 + b1[f];
        B[(size_t)i * HID + f] = v > 0.0f ? v : 0.0f;
    }
}

// -------------------------------- layer-2 epilogue: self-loop+bias+log_softmax
__global__ __launch_bounds__(256) void logsoftmax_out(float* __restrict__ out,
                                                      const float* __restrict__ P,
                                                      const float* __restrict__ dinv,
                                                      const float* __restrict__ b2,
                                                      int n) {
    int i = blockIdx.x * blockDim.x + threadIdx.x;
    if (i >= n) return;
    const float d2 = dinv[i] * dinv[i];
    float v[NCLS];
    float m = -3.4e38f;
#pragma unroll
    for (int f = 0; f < NCLS; ++f) {
        v[f] = out[(size_t)i * NCLS + f] + P[(size_t)i * 16 + f] * d2 + b2[f];
        m = fmaxf(m, v[f]);
    }
    float s = 0.0f;
#pragma unroll
    for (int f = 0; f < NCLS; ++f) s += __expf(v[f] - m);
    const float lse = m + __logf(s);
#pragma unroll
    for (int f = 0; f < NCLS; ++f) out[(size_t)i * NCLS + f] = v[f] - lse;
}

// -------------------------------------------------------------------- driver
extern "C" void kernel_launch(void* const* d_in, const int* in_sizes, int n_in,
                              void* d_out, int out_size, void* d_ws, size_t ws_size,
                              hipStream_t stream) {
    const float* x   = (const float*)d_in[0];
    const int*   ei  = (const int*)d_in[1];
    const float* W1  = (const float*)d_in[2];
    const float* b1  = (const float*)d_in[3];
    const float* W2  = (const float*)d_in[4];
    const float* b2  = (const float*)d_in[5];
    float*       out = (float*)d_out;

    const int n = in_sizes[0] / F_IN;     // 100000
    const int e = in_sizes[1] / 2;        // 3200000
    const int* srcIdx = ei;               // edge_index[0]
    const int* dstIdx = ei + e;           // edge_index[1]

    float* dinv = (float*)d_ws;           // N
    float* A    = dinv + n;               // N*16 : projections
    float* B    = A + (size_t)n * HID;    // N*16 : aggregation / hidden

    const int nTiles = (n + 15) / 16;     // N % 16 == 0 here
    const int nodeBlk = (n + 255) / 256;
    const int edgeBlk = (e + 255) / 256;

    // 1) degrees -> dinv
    zero_f32<<<nodeBlk, 256, 0, stream>>>(dinv, n);
    deg_count<<<edgeBlk, 256, 0, stream>>>(dstIdx, dinv, e);
    finalize_dinv<<<nodeBlk, 256, 0, stream>>>(dinv, n);

    // 2) layer 1: project, scatter, epilogue
    gemm_x_w1<<<(nTiles + 7) / 8, 256, 0, stream>>>(x, W1, A, nTiles);
    zero_f32<<<(n * HID + 255) / 256, 256, 0, stream>>>(B, n * HID);
    scatter_edges<HID, HID, HID><<<edgeBlk, 256, 0, stream>>>(srcIdx, dstIdx, dinv, A, B, e);
    relu_bias_self<<<nodeBlk, 256, 0, stream>>>(A, B, dinv, b1, n);

    // 3) layer 2: project (padded to 16), scatter into out, epilogue in-place
    gemm_h_w2<<<(nTiles + 7) / 8, 256, 0, stream>>>(B, W2, A, nTiles);
    zero_f32<<<(n * NCLS + 255) / 256, 256, 0, stream>>>(out, n * NCLS);
    scatter_edges<NCLS, 16, NCLS><<<edgeBlk, 256, 0, stream>>>(srcIdx, dstIdx, dinv, A, out, e);
    logsoftmax_out<<<nodeBlk, 256, 0, stream>>>(out, A, dinv, b2, n);
}